// autoencoder_35064113004837
// MI455X (gfx1250) — compile-verified
//
#include <hip/hip_runtime.h>
#include <math.h>

typedef __attribute__((ext_vector_type(2))) float v2f;
typedef __attribute__((ext_vector_type(8))) float v8f;

#define NN     4096
#define NF     1024
#define NHID   64
#define NHEADS 8
#define EMBD   512
#define LALPHA 0.2f

__device__ __forceinline__ v8f wmma_f32(v2f a, v2f b, v8f c) {
  // D = A(16x4 f32) x B(4x16 f32) + C(16x16 f32)
  return __builtin_amdgcn_wmma_f32_16x16x4_f32(false, a, false, b, (short)0, c,
                                               false, false);
}

// Async global->LDS copy, 16B per lane (ASYNCcnt-tracked, no VGPR round trip).
__device__ __forceinline__ void async_ld_b128(unsigned lds_off, int gvoff,
                                              const void* base) {
  asm volatile("global_load_async_to_lds_b128 %0, %1, %2"
               :: "v"(lds_off), "v"(gvoff), "s"(base) : "memory");
}
// 8B-per-lane variant.
__device__ __forceinline__ void async_ld_b64(unsigned lds_off, int gvoff,
                                             const void* base) {
  asm volatile("global_load_async_to_lds_b64 %0, %1, %2"
               :: "v"(lds_off), "v"(gvoff), "s"(base) : "memory");
}
__device__ __forceinline__ unsigned lds_addr(const void* p) {
  return (unsigned)(size_t)p;  // addrspace(3) pointer == LDS byte offset
}

// C[M,N] = act(A[M,K] @ op(B) + bias), op(B)=B^T if BT (B stored [N,K]) else B [K,N].
// Block: 256 threads (8 waves). Tile: BM=128 (16 rows/wave), BN=64, KC=16.
// LDS tiles double-buffered, filled via async-to-LDS; pads keep 16B chunk
// alignment (20*4=80, 68*4=272 bytes) and conflict-free fragment reads.
template <bool BT, bool RELU, bool HASBIAS>
__global__ __launch_bounds__(256) void k_gemm(
    const float* __restrict__ A, const float* __restrict__ B,
    const float* __restrict__ bias, float* __restrict__ C, int M, int N, int K,
    int lda, int ldb, int ldc, long long strideB, long long strideC) {
  (void)M; (void)N;
  __shared__ float As[2][128 * 20];   // [row][k] pad 20
  __shared__ float Bsh[2][1280];      // BT: [n][k] pad 20 ; !BT: [k][n] pad 68
  const int tid = threadIdx.x;
  const int wv = tid >> 5;
  const int lane = tid & 31;
  const int ln = lane & 15;
  const int hi = lane >> 4;
  const int row0 = blockIdx.x * 128;
  const int col0 = blockIdx.y * 64;
  const float* Bp = B + (long long)blockIdx.z * strideB;
  float* Cp = C + (long long)blockIdx.z * strideC;

  v8f acc[4] = {};

  // Issue async fills of one K-tile into buffer `buf` (3 instr per wave).
  auto issue_tile = [&](int buf, int k0) {
#pragma unroll
    for (int i = 0; i < 2; ++i) {  // A: 128x16 f32 = 512 x 16B chunks
      int c = tid + i * 256;
      int r = c >> 2, c4 = c & 3;
      async_ld_b128(lds_addr(&As[buf][r * 20 + c4 * 4]),
                    (int)(((row0 + r) * (long long)lda + k0 + c4 * 4) * 4), A);
    }
    {  // B: 16x64 f32 = 256 x 16B chunks (1 per thread)
      int c = tid;
      if (BT) {
        int n = c >> 2, k4 = c & 3;
        async_ld_b128(lds_addr(&Bsh[buf][n * 20 + k4 * 4]),
                      (int)(((col0 + n) * (long long)ldb + k0 + k4 * 4) * 4),
                      Bp);
      } else {
        int k = c >> 4, n4 = c & 15;
        async_ld_b128(lds_addr(&Bsh[buf][k * 68 + n4 * 4]),
                      (int)(((k0 + k) * (long long)ldb + col0 + n4 * 4) * 4),
                      Bp);
      }
    }
  };

  const int nk = K >> 4;
  issue_tile(0, 0);
  for (int kt = 0; kt < nk; ++kt) {
    const int cur = kt & 1;
    __syncthreads();  // prior compute done: safe to overwrite buffer cur^1
    if (kt + 1 < nk) {
      issue_tile(cur ^ 1, (kt + 1) << 4);
      asm volatile("s_wait_asynccnt 0x3" ::: "memory");  // tile kt landed
    } else {
      asm volatile("s_wait_asynccnt 0x0" ::: "memory");
    }
    __syncthreads();  // tile kt visible to all waves
#pragma unroll
    for (int kk = 0; kk < 4; ++kk) {
      const int kb = kk * 4 + hi * 2;
      v2f a;  // A-frag: lane<16 -> K=kb,kb+1 ; lane>=16 -> +2
      a.x = As[cur][(wv * 16 + ln) * 20 + kb];
      a.y = As[cur][(wv * 16 + ln) * 20 + kb + 1];
#pragma unroll
      for (int ns = 0; ns < 4; ++ns) {
        v2f b;
        if (BT) {
          b.x = Bsh[cur][(ns * 16 + ln) * 20 + kb];
          b.y = Bsh[cur][(ns * 16 + ln) * 20 + kb + 1];
        } else {
          b.x = Bsh[cur][kb * 68 + ns * 16 + ln];
          b.y = Bsh[cur][(kb + 1) * 68 + ns * 16 + ln];
        }
        acc[ns] = wmma_f32(a, b, acc[ns]);
      }
    }
  }
#pragma unroll
  for (int ns = 0; ns < 4; ++ns) {
#pragma unroll
    for (int r = 0; r < 8; ++r) {
      int row = row0 + wv * 16 + (hi ? 8 : 0) + r;
      int col = col0 + ns * 16 + ln;
      float v = acc[ns][r];
      if (HASBIAS) v += bias[col];
      if (RELU) v = fmaxf(v, 0.f);
      Cp[(long long)row * ldc + col] = v;
    }
  }
}

// e_src[h,n] = Wh[h,n,:].a1[h]; e_dst likewise with a2. One thread per (h,n).
__global__ __launch_bounds__(256) void k_edge(const float* __restrict__ Wh,
                                              const float* __restrict__ a1,
                                              const float* __restrict__ a2,
                                              float* __restrict__ es,
                                              float* __restrict__ ed) {
  int t = blockIdx.x * 256 + threadIdx.x;  // t = h*4096 + n
  int h = t >> 12;
  const float4* w = (const float4*)(Wh + (long long)t * NHID);
  const float4* p1 = (const float4*)(a1 + h * NHID);
  const float4* p2 = (const float4*)(a2 + h * NHID);
  float s1 = 0.f, s2 = 0.f;
#pragma unroll
  for (int i = 0; i < 16; ++i) {
    float4 wv = w[i], x1 = p1[i], x2 = p2[i];
    s1 += wv.x * x1.x + wv.y * x1.y + wv.z * x1.z + wv.w * x1.w;
    s2 += wv.x * x2.x + wv.y * x2.y + wv.z * x2.z + wv.w * x2.w;
  }
  es[t] = s1;
  ed[t] = s2;
}

// One wave per row n: online masked softmax stats for all 8 heads at once.
__global__ __launch_bounds__(256) void k_rowstats(
    const float* __restrict__ adj, const float* __restrict__ es,
    const float* __restrict__ ed, float* __restrict__ rmax,
    float* __restrict__ rinv) {
  const int wv = threadIdx.x >> 5, lane = threadIdx.x & 31;
  const int n = blockIdx.x * 8 + wv;
  float esv[NHEADS], mx[NHEADS], sm[NHEADS];
#pragma unroll
  for (int h = 0; h < NHEADS; ++h) {
    esv[h] = es[h * NN + n];
    mx[h] = -1e30f;
    sm[h] = 0.f;
  }
  const float* arow = adj + (long long)n * NN;
  for (int m = lane; m < NN; m += 32) {
    float a = arow[m];
    if (a > 0.f) {  // ~1% taken: skip 8-head update otherwise
#pragma unroll
      for (int h = 0; h < NHEADS; ++h) {
        float e = esv[h] + ed[h * NN + m];
        e = e > 0.f ? e : LALPHA * e;
        float nm = fmaxf(mx[h], e);
        sm[h] = sm[h] * __expf(mx[h] - nm) + __expf(e - nm);
        mx[h] = nm;
      }
    }
  }
#pragma unroll
  for (int off = 16; off >= 1; off >>= 1) {
#pragma unroll
    for (int h = 0; h < NHEADS; ++h) {
      float omx = __shfl_xor(mx[h], off, 32);
      float osm = __shfl_xor(sm[h], off, 32);
      float nm = fmaxf(mx[h], omx);
      sm[h] = sm[h] * __expf(mx[h] - nm) + osm * __expf(omx - nm);
      mx[h] = nm;
    }
  }
  if (lane == 0) {
#pragma unroll
    for (int h = 0; h < NHEADS; ++h) {
      rmax[h * NN + n] = mx[h];
      rinv[h * NN + n] = 1.f / sm[h];
    }
  }
}

// enc = elu(softmax(masked e) @ Wh), interleaved [N, H*64].
// Block = 8 waves; wave w = head w; 32 rows/block; adj tile async-staged in
// LDS (double buffered, pad 18 words -> 8B-aligned b64 chunks).
__global__ __launch_bounds__(256) void k_attn(
    const float* __restrict__ adj, const float* __restrict__ Wh,
    const float* __restrict__ es, const float* __restrict__ ed,
    const float* __restrict__ rmax, const float* __restrict__ rinv,
    float* __restrict__ enc) {
  __shared__ float adjt[2][32 * 18];
  const int tid = threadIdx.x;
  const int h = tid >> 5;
  const int lane = tid & 31, ln = lane & 15, hi = lane >> 4;
  const int r0 = blockIdx.x * 32;
  const float* Whh = Wh + (long long)h * NN * NHID;
  const float es0 = es[h * NN + r0 + ln];
  const float es1 = es[h * NN + r0 + 16 + ln];
  const float rm0 = rmax[h * NN + r0 + ln];
  const float rm1 = rmax[h * NN + r0 + 16 + ln];
  v8f acc0[4] = {}, acc1[4] = {};

  auto issue_adj = [&](int buf, int m0) {  // 32x16 f32 = 256 x 8B chunks
    int r = tid >> 3, c2 = tid & 7;
    async_ld_b64(lds_addr(&adjt[buf][r * 18 + c2 * 2]),
                 (int)(((r0 + r) * (long long)NN + m0 + c2 * 2) * 4), adj);
  };

  issue_adj(0, 0);
  for (int mt = 0; mt < NN / 16; ++mt) {
    const int m0 = mt << 4;
    const int cur = mt & 1;
    __syncthreads();  // prior tile fully consumed
    if (mt + 1 < NN / 16) {
      issue_adj(cur ^ 1, m0 + 16);
      asm volatile("s_wait_asynccnt 0x1" ::: "memory");
    } else {
      asm volatile("s_wait_asynccnt 0x0" ::: "memory");
    }
    __syncthreads();  // tile mt visible to all waves
#pragma unroll
    for (int kk = 0; kk < 4; ++kk) {
      const int kb = kk * 4 + hi * 2;
      float ed0 = ed[h * NN + m0 + kb];
      float ed1 = ed[h * NN + m0 + kb + 1];
      v2f a0, a1;  // probability tiles built directly in A-frag layout
      {
        float e = es0 + ed0; e = e > 0.f ? e : LALPHA * e;
        a0.x = (adjt[cur][ln * 18 + kb] > 0.f) ? __expf(e - rm0) : 0.f;
        e = es0 + ed1; e = e > 0.f ? e : LALPHA * e;
        a0.y = (adjt[cur][ln * 18 + kb + 1] > 0.f) ? __expf(e - rm0) : 0.f;
        e = es1 + ed0; e = e > 0.f ? e : LALPHA * e;
        a1.x = (adjt[cur][(16 + ln) * 18 + kb] > 0.f) ? __expf(e - rm1) : 0.f;
        e = es1 + ed1; e = e > 0.f ? e : LALPHA * e;
        a1.y = (adjt[cur][(16 + ln) * 18 + kb + 1] > 0.f) ? __expf(e - rm1) : 0.f;
      }
      const float* wp = Whh + (long long)(m0 + kb) * NHID;
#pragma unroll
      for (int ns = 0; ns < 4; ++ns) {
        v2f b;  // Wh is [m, d] row-major: native B layout
        b.x = wp[ns * 16 + ln];
        b.y = wp[NHID + ns * 16 + ln];
        acc0[ns] = wmma_f32(a0, b, acc0[ns]);
        acc1[ns] = wmma_f32(a1, b, acc1[ns]);
      }
    }
  }
#pragma unroll
  for (int r = 0; r < 8; ++r) {
    int rowA = r0 + (hi ? 8 : 0) + r;
    int rowB = rowA + 16;
    float i0 = rinv[h * NN + rowA];
    float i1 = rinv[h * NN + rowB];
#pragma unroll
    for (int ns = 0; ns < 4; ++ns) {
      int col = h * NHID + ns * 16 + ln;
      float v = acc0[ns][r] * i0;
      v = v > 0.f ? v : __expf(v) - 1.f;  // elu
      enc[(long long)rowA * EMBD + col] = v;
      v = acc1[ns][r] * i1;
      v = v > 0.f ? v : __expf(v) - 1.f;
      enc[(long long)rowB * EMBD + col] = v;
    }
  }
}

extern "C" void kernel_launch(void* const* d_in, const int* in_sizes, int n_in,
                              void* d_out, int out_size, void* d_ws,
                              size_t ws_size, hipStream_t stream) {
  (void)in_sizes; (void)n_in; (void)out_size; (void)ws_size;
  const float* x   = (const float*)d_in[0];
  const float* adj = (const float*)d_in[1];
  const float* W   = (const float*)d_in[2];
  const float* a1  = (const float*)d_in[3];
  const float* a2  = (const float*)d_in[4];
  const float* W1  = (const float*)d_in[5];
  const float* b1  = (const float*)d_in[6];
  const float* W2  = (const float*)d_in[7];
  const float* b2  = (const float*)d_in[8];
  const float* W3  = (const float*)d_in[9];
  const float* b3  = (const float*)d_in[10];
  const float* W4  = (const float*)d_in[11];
  const float* b4  = (const float*)d_in[12];
  float* out = (float*)d_out;

  float* ws = (float*)d_ws;
  float* Wh   = ws; ws += (size_t)NHEADS * NN * NHID;  // 8 MB
  float* esrc = ws; ws += (size_t)NHEADS * NN;
  float* edst = ws; ws += (size_t)NHEADS * NN;
  float* rmax = ws; ws += (size_t)NHEADS * NN;
  float* rinv = ws; ws += (size_t)NHEADS * NN;
  float* enc  = ws; ws += (size_t)NN * EMBD;           // 8 MB
  float* h1   = ws; ws += (size_t)NN * 256;
  float* h2   = ws; ws += (size_t)NN * 256;
  float* h3   = ws; ws += (size_t)NN * 512;

  // 1) Wh[h] = x @ W[h]   (batched over heads via grid.z)
  k_gemm<false, false, false><<<dim3(NN / 128, 1, NHEADS), 256, 0, stream>>>(
      x, W, nullptr, Wh, NN, NHID, NF, NF, NHID, NHID,
      (long long)NF * NHID, (long long)NN * NHID);
  // 2) attention logits per node
  k_edge<<<dim3(NHEADS * NN / 256), 256, 0, stream>>>(Wh, a1, a2, esrc, edst);
  // 3) masked-softmax row stats (max, 1/sum)
  k_rowstats<<<dim3(NN / 8), 256, 0, stream>>>(adj, esrc, edst, rmax, rinv);
  // 4) enc = elu(att @ Wh), concat heads
  k_attn<<<dim3(NN / 32), 256, 0, stream>>>(adj, Wh, esrc, edst, rmax, rinv, enc);
  // 5-8) decoder MLP (y = x @ W^T + b)
  k_gemm<true, true, true><<<dim3(NN / 128, 256 / 64), 256, 0, stream>>>(
      enc, W1, b1, h1, NN, 256, EMBD, EMBD, EMBD, 256, 0, 0);
  k_gemm<true, true, true><<<dim3(NN / 128, 256 / 64), 256, 0, stream>>>(
      h1, W2, b2, h2, NN, 256, 256, 256, 256, 256, 0, 0);
  k_gemm<true, true, true><<<dim3(NN / 128, 512 / 64), 256, 0, stream>>>(
      h2, W3, b3, h3, NN, 512, 256, 256, 256, 512, 0, 0);
  k_gemm<true, false, true><<<dim3(NN / 128, 1024 / 64), 256, 0, stream>>>(
      h3, W4, b4, out, NN, 1024, 512, 512, 512, 1024, 0, 0);
}